// TorchBiosystemModelWithInputs_62809601737326
// MI455X (gfx1250) — compile-verified
//
#include <hip/hip_runtime.h>
#include <stdint.h>

// Problem constants (from reference): B=65536 trajectories, T=512 time points.
#define NBTRAJ 65536
#define NSTEPS 512
#define TPB    256

typedef float v2f __attribute__((ext_vector_type(2)));
typedef float v4f __attribute__((ext_vector_type(4)));

__device__ __forceinline__ v2f splat2(float x) { v2f v; v.x = x; v.y = x; return v; }

// Fast hardware reciprocal (v_rcp_f32, ~1 ulp) instead of full IEEE divide expansion.
__device__ __forceinline__ v2f rcp2(v2f x) {
    v2f r;
    r.x = __builtin_amdgcn_rcpf(x.x);
    r.y = __builtin_amdgcn_rcpf(x.y);
    return r;
}

__device__ __forceinline__ v2f fma2(v2f a, v2f b, v2f c) {
    return __builtin_elementwise_fma(a, b, c);
}

// CDNA5 async global->LDS copy (tracked by ASYNCcnt). VDST VGPR holds the LDS
// byte address (wave-LDS-base relative == low 32 bits of the generic pointer).
__device__ __forceinline__ void async_ld_b32(uint32_t lds_off, const float* g) {
    asm volatile("global_load_async_to_lds_b32 %0, %1, off"
                 :: "v"(lds_off), "v"((uint64_t)(uintptr_t)g)
                 : "memory");
}

__device__ __forceinline__ void wait_async0() {
    asm volatile("s_wait_asynccnt 0" ::: "memory");
}

// rhs for two independent trajectories packed as v2f:
//   rho  = mu*s / (Km + s)
//   db   = -D*b + rho*b
//   ds   =  D*(se - s) - rho*b
__device__ __forceinline__ void rhsv(v2f b, v2f s, float D, float Km, float se, float mu,
                                     v2f& db, v2f& ds) {
    v2f inv = rcp2(splat2(Km) + s);
    v2f rho = (mu * s) * inv;
    v2f rb  = rho * b;
    db = fma2(splat2(-D), b, rb);
    ds = fma2(splat2(D), splat2(se) - s, -rb);
}

__global__ __launch_bounds__(TPB)
void rk4_chemostat_kernel(const float* __restrict__ y0,
                          const float* __restrict__ t_eval,
                          const float* __restrict__ D_seq,
                          const float* __restrict__ pKm,
                          const float* __restrict__ pSe,
                          const float* __restrict__ pMu,
                          float* __restrict__ out) {
    __shared__ float sh_t[NSTEPS];
    __shared__ float sh_dt[NSTEPS - 1];
    __shared__ float sh_D[NSTEPS - 1];

    const int tid = threadIdx.x;

    // ---- async stage t_eval[512] and D_seq[511] into LDS ----
    async_ld_b32((uint32_t)(uintptr_t)&sh_t[tid],        t_eval + tid);
    async_ld_b32((uint32_t)(uintptr_t)&sh_t[tid + TPB],  t_eval + tid + TPB);
    async_ld_b32((uint32_t)(uintptr_t)&sh_D[tid],        D_seq + tid);
    if (tid < (NSTEPS - 1 - TPB)) {
        async_ld_b32((uint32_t)(uintptr_t)&sh_D[tid + TPB], D_seq + tid + TPB);
    }

    // Uniform model parameters (single-element inputs).
    const float Km = *pKm;
    const float se = *pSe;
    const float mu = *pMu;

    wait_async0();
    __syncthreads();

    // Build dt[i] = t[i+1] - t[i] in LDS.
    sh_dt[tid] = sh_t[tid + 1] - sh_t[tid];
    if (tid < (NSTEPS - 1 - TPB)) {
        sh_dt[tid + TPB] = sh_t[tid + TPB + 1] - sh_t[tid + TPB];
    }
    __syncthreads();

    // Each thread owns ADJACENT trajectories 2*gid and 2*gid+1:
    //  - packed f32 math across the pair (v_pk_fma_f32 / dual-issue)
    //  - one contiguous 16B NT store per step (global_store_b128, 512B/wave burst)
    const int gid = blockIdx.x * TPB + tid;   // 0 .. NBTRAJ/2 - 1

    const v4f* y0v  = (const v4f*)y0;         // {b0,s0,b1,s1} per pair
    v4f*       outv = (v4f*)out;              // row stride = NBTRAJ/2 v4f elements

    v4f y = y0v[gid];

    // Row 0 of output is y0 (non-temporal: output is written once, 256MB total).
    __builtin_nontemporal_store(y, &outv[gid]);

    v2f vb = { y.x, y.z };   // b-state of both trajectories
    v2f vs = { y.y, y.w };   // s-state of both trajectories

    for (int i = 0; i < NSTEPS - 1; ++i) {
        const float dt = sh_dt[i];   // LDS broadcast reads (all lanes same addr)
        const float D  = sh_D[i];
        const float h2 = 0.5f * dt;
        const float h6 = dt * 0.16666666666666666f;

        v2f k1b, k1s, k2b, k2s, k3b, k3s, k4b, k4s;
        rhsv(vb, vs, D, Km, se, mu, k1b, k1s);
        rhsv(fma2(splat2(h2), k1b, vb), fma2(splat2(h2), k1s, vs), D, Km, se, mu, k2b, k2s);
        rhsv(fma2(splat2(h2), k2b, vb), fma2(splat2(h2), k2s, vs), D, Km, se, mu, k3b, k3s);
        rhsv(fma2(splat2(dt), k3b, vb), fma2(splat2(dt), k3s, vs), D, Km, se, mu, k4b, k4s);

        v2f sb = k1b + 2.0f * k2b + 2.0f * k3b + k4b;
        v2f ss = k1s + 2.0f * k2s + 2.0f * k3s + k4s;
        vb = fma2(splat2(h6), sb, vb);
        vs = fma2(splat2(h6), ss, vs);

        const size_t row = (size_t)(i + 1) * (NBTRAJ / 2);
        v4f o = { vb.x, vs.x, vb.y, vs.y };
        __builtin_nontemporal_store(o, &outv[row + gid]);
    }
}

extern "C" void kernel_launch(void* const* d_in, const int* in_sizes, int n_in,
                              void* d_out, int out_size, void* d_ws, size_t ws_size,
                              hipStream_t stream) {
    (void)in_sizes; (void)n_in; (void)out_size; (void)d_ws; (void)ws_size;

    const float* y0     = (const float*)d_in[0];  // (B, 2)
    const float* t_eval = (const float*)d_in[1];  // (T,)
    const float* D_seq  = (const float*)d_in[2];  // (T-1,)
    const float* Km     = (const float*)d_in[3];  // scalar
    const float* se     = (const float*)d_in[4];  // scalar
    const float* mu     = (const float*)d_in[5];  // scalar
    float*       out    = (float*)d_out;          // (T, B, 2)

    const int nblocks = NBTRAJ / (TPB * 2);       // 128 blocks, 2 trajectories/thread
    rk4_chemostat_kernel<<<nblocks, TPB, 0, stream>>>(y0, t_eval, D_seq, Km, se, mu, out);
}